// DWTLayer_69750268887081
// MI455X (gfx1250) — compile-verified
//
#include <hip/hip_runtime.h>
#include <stdint.h>

#ifndef __has_builtin
#define __has_builtin(x) 0
#endif

#define T_LEN   16384
#define C_CH    64
#define M_OUT   8190                 // T/2 - 2
#define M_TILE  64
#define TILE_ROWS (2*M_TILE + 7)     // 135 input rows per tile
#define THREADS 256

typedef unsigned int v4u __attribute__((ext_vector_type(4)));
typedef int          v8i __attribute__((ext_vector_type(8)));
typedef int          v4i __attribute__((ext_vector_type(4)));

__global__ __launch_bounds__(THREADS)
void dwt_db4_tdm_kernel(const float* __restrict__ x,
                        const float* __restrict__ dec_lo,
                        const float* __restrict__ dec_hi,
                        float* __restrict__ out)
{
    __shared__ float tile[TILE_ROWS * C_CH];    // 34,560 B

    const int tilesPerB = (M_OUT + M_TILE - 1) / M_TILE;   // 128
    const int b  = (int)blockIdx.x / tilesPerB;
    const int tb = (int)blockIdx.x - b * tilesPerB;
    const int m0 = tb * M_TILE;

    // input row span: r0 .. r0+134  (global rows, pre-reflection)
    const int r0    = 2*m0 - 1;
    const int rbase = (r0 < 0) ? 0 : r0;
    int rlast = 2*m0 + 2*M_TILE + 5;            // r0 + 134
    if (rlast > T_LEN - 1) rlast = T_LEN - 1;
    const int nrows = rlast - rbase + 1;
    const unsigned nelem = (unsigned)(nrows * C_CH);   // <= 8640 elements

    const float* gsrc = x + ((size_t)b * T_LEN + (size_t)rbase) * C_CH;

#if __has_builtin(__builtin_amdgcn_tensor_load_to_lds) && __has_builtin(__builtin_amdgcn_s_wait_tensorcnt)
    // --- Tensor Data Mover: one 1-D DMA descriptor copies the whole tile ---
    if (threadIdx.x < 32) {                     // wave 0 only (TDM ignores EXEC;
                                                //  scalar branch keeps other waves out)
        const unsigned long long gaddr = (unsigned long long)(uintptr_t)gsrc;
        const unsigned lds_off = (unsigned)(uintptr_t)&tile[0];  // low 32b = LDS byte offset

        v4u g0;                                  // D# group 0 (128b)
        g0[0] = 1u;                              // count=1 (valid descriptor)
        g0[1] = lds_off;                         // lds_addr [63:32]
        g0[2] = (unsigned)(gaddr & 0xffffffffull);              // global_addr[31:0]
        g0[3] = (unsigned)((gaddr >> 32) & 0x01ffffffull)       // global_addr[56:32]
              | (2u << 30);                      // type=2 ("image")

        v8i g1;                                  // D# group 1 (256b)
        g1[0] = (int)(2u << 16);                 // wg_mask=0, data_size=2 (4B)
        g1[1] = (int)((nelem & 0xffffu) << 16);  // tensor_dim0[15:0] at bits 63:48
        g1[2] = (int)(((nelem >> 16) & 0xffffu)  // tensor_dim0[31:16]
              | (1u << 16));                     // tensor_dim1 = 1
        g1[3] = (int)((nelem & 0xffffu) << 16);  // tile_dim0 = nelem (bits 127:112)
        g1[4] = 1;                               // tile_dim1 = 1, tile_dim2 = 0
        g1[5] = (int)nelem;                      // tensor_dim0_stride (low 32)
        g1[6] = 0;
        g1[7] = 0;

        v4i z4; z4[0] = 0; z4[1] = 0; z4[2] = 0; z4[3] = 0;   // groups 2/3 unused (<=2D)
        v8i z8; z8[0]=0; z8[1]=0; z8[2]=0; z8[3]=0; z8[4]=0; z8[5]=0; z8[6]=0; z8[7]=0;

        // 6-arg form (clang-23 / therock-10.0 headers lane)
        __builtin_amdgcn_tensor_load_to_lds(g0, g1, z4, z4, z8, 0);
        __builtin_amdgcn_s_wait_tensorcnt(0);
    }
    __syncthreads();
#else
    // Fallback: cooperative vectorized copy (tile is contiguous in memory)
    {
        const float4* s4 = (const float4*)gsrc;
        float4* d4 = (float4*)tile;
        const unsigned n4 = nelem >> 2;
        for (unsigned i = threadIdx.x; i < n4; i += THREADS) d4[i] = s4[i];
    }
    __syncthreads();
#endif

    // filter taps: uniform addresses -> scalar loads
    float flo[8], fhi[8];
#pragma unroll
    for (int k = 0; k < 8; ++k) { flo[k] = dec_lo[k]; fhi[k] = dec_hi[k]; }

    const int mt = threadIdx.x >> 4;          // 0..15 : group of 4 output rows
    const int c4 = (threadIdx.x & 15) << 2;   // float4 channel offset (0..60)
    const int m_first = m0 + mt * 4;

    // sliding window: 15 LDS rows cover outputs m_first .. m_first+3
    float4 w[15];
#pragma unroll
    for (int t = 0; t < 15; ++t) {
        int r  = 2*m_first - 1 + t;                                   // global row
        int rr = (r < 0) ? -r : ((r > T_LEN-1) ? (2*(T_LEN-1) - r) : r); // reflect
        w[t] = *(const float4*)&tile[(rr - rbase) * C_CH + c4];
    }

    float* outb = out + (size_t)b * M_OUT * (2*C_CH);
#pragma unroll
    for (int j = 0; j < 4; ++j) {
        const int m = m_first + j;
        float ax=0.f, ay=0.f, az=0.f, aw=0.f;
        float dx=0.f, dy=0.f, dz=0.f, dw=0.f;
#pragma unroll
        for (int k = 0; k < 8; ++k) {
            const float4 v = w[2*j + k];
            ax = fmaf(v.x, flo[k], ax); ay = fmaf(v.y, flo[k], ay);
            az = fmaf(v.z, flo[k], az); aw = fmaf(v.w, flo[k], aw);
            dx = fmaf(v.x, fhi[k], dx); dy = fmaf(v.y, fhi[k], dy);
            dz = fmaf(v.z, fhi[k], dz); dw = fmaf(v.w, fhi[k], dw);
        }
        if (m < M_OUT) {
            float4 a; a.x = ax; a.y = ay; a.z = az; a.w = aw;
            float4 d; d.x = dx; d.y = dy; d.z = dz; d.w = dw;
            float* o = outb + (size_t)m * (2*C_CH);
            *(float4*)&o[c4]        = a;   // approx -> channels [0,64)
            *(float4*)&o[C_CH + c4] = d;   // detail -> channels [64,128)
        }
    }
}

extern "C" void kernel_launch(void* const* d_in, const int* in_sizes, int n_in,
                              void* d_out, int out_size, void* d_ws, size_t ws_size,
                              hipStream_t stream)
{
    const float* x  = (const float*)d_in[0];
    const float* lo = (const float*)d_in[1];
    const float* hi = (const float*)d_in[2];
    float* out = (float*)d_out;

    const int B = in_sizes[0] / (T_LEN * C_CH);             // 32
    const int tilesPerB = (M_OUT + M_TILE - 1) / M_TILE;    // 128

    dim3 grid((unsigned)(B * tilesPerB)), block(THREADS);
    hipLaunchKernelGGL(dwt_db4_tdm_kernel, grid, block, 0, stream, x, lo, hi, out);
}